// wBCE_loss_41618233098518
// MI455X (gfx1250) — compile-verified
//
#include <hip/hip_runtime.h>
#include <hip/hip_bf16.h>
#include <math.h>

// ---------------------------------------------------------------------------
// Fused 31x31 avg-pool + weighted BCE-with-logits loss for MI455X (gfx1250).
// Horizontal 31-tap box sum done on the WMMA pipe as banded 16x16 matmuls
// (V_WMMA_F32_16X16X4_F32), vertical 31-tap sum as LDS sliding-window.
// Memory-bound problem: ~106 MB traffic -> ~4.5 us at 23.3 TB/s.
// ---------------------------------------------------------------------------

typedef __attribute__((ext_vector_type(2))) float v2f;
typedef __attribute__((ext_vector_type(8))) float v8f;

#define IMG_H 512
#define IMG_W 512
#define N_IMG 32
#define TILE 64
#define HALO 15
#define HR (TILE + 2 * HALO)   // 94 halo rows/cols
#define HCST 96                // halo LDS col stride (padded)
#define VCST 96                // vsum LDS cols: logical [-16, 79] -> storage [0, 95]

__global__ void wbce_init_acc(float* acc) {
    if (threadIdx.x < 2 * N_IMG) acc[threadIdx.x] = 0.0f;
}

__global__ __launch_bounds__(512) void wbce_pool_kernel(
    const float* __restrict__ logits,
    const float* __restrict__ targets,
    float* __restrict__ acc) {

    __shared__ float sT[HR][HCST];    // targets halo: 94x96 = 36 KB
    __shared__ float sV[TILE][VCST];  // vertical 31-sums: 64x96 = 24 KB

    const int b    = blockIdx.x >> 6;        // image
    const int tile = blockIdx.x & 63;        // 8x8 tiles of 64x64
    const int h0   = (tile >> 3) * TILE;
    const int w0   = (tile & 7) * TILE;
    const int tid  = threadIdx.x;

    const float* timg = targets + (size_t)b * IMG_H * IMG_W;
    const float* limg = logits  + (size_t)b * IMG_H * IMG_W;

    // ---- Phase 1: load zero-padded targets halo (count_include_pad => 0-pad)
    for (int i = tid; i < HR * HR; i += 512) {
        int hr = i / HR, hc = i % HR;
        int h = h0 - HALO + hr;
        int w = w0 - HALO + hc;
        float v = 0.0f;
        if ((unsigned)h < IMG_H && (unsigned)w < IMG_W) v = timg[h * IMG_W + w];
        sT[hr][hc] = v;
    }
    __syncthreads();

    // ---- Phase 2: vertical sliding-window 31-sums (with horizontal halo)
    // Guard columns (logical -16 and +79) get weight 0 in the band matmul,
    // but must not hold NaN garbage: zero them explicitly.
    if (tid < TILE) { sV[tid][0] = 0.0f; sV[tid][VCST - 1] = 0.0f; }
    if (tid < HR * 4) {                 // 376 active threads
        int c  = tid % HR;              // halo col 0..93 (logical col c-15)
        int rg = tid / HR;              // row group of 16
        int r0 = rg * 16;
        float s = 0.0f;
        #pragma unroll
        for (int dy = 0; dy < 31; ++dy) s += sT[r0 + dy][c];
        sV[r0][c + 1] = s;              // storage col = logical + 16 = c + 1
        #pragma unroll
        for (int r = 1; r < 16; ++r) {
            s += sT[r0 + r + 30][c] - sT[r0 + r - 1][c];
            sV[r0 + r][c + 1] = s;
        }
    }
    __syncthreads();

    // ---- Phase 3: horizontal 31-tap sum as banded WMMA matmuls.
    // Out = V(-1)*B(-1) + V(0)*ones + V(+1)*B(+1), B(-1)[k][n]=(k>n), B(+1)=(k<n).
    const int wv   = tid >> 5;          // wave 0..15, one 16x16 subtile each
    const int lane = tid & 31;
    const int sr   = wv >> 2;           // subtile row 0..3
    const int sc   = wv & 3;            // subtile col 0..3
    const int m    = lane & 15;         // A: M index; B/D: N index
    const int hi   = lane >> 4;         // upper half-wave

    v8f d = {};
    #pragma unroll
    for (int o = 0; o < 3; ++o) {       // column-tile offset: o-1 in {-1,0,+1}
        #pragma unroll
        for (int kk = 0; kk < 4; ++kk) {
            const int kbase = kk * 4 + hi * 2;   // K indices for this lane half
            // A fragment: vsum[16*sr + m][logical col 16*sc + 16*(o-1) + k]
            const int scol = 16 + sc * 16 + (o - 1) * 16 + kbase; // storage col
            v2f a;
            a.x = sV[sr * 16 + m][scol];
            a.y = sV[sr * 16 + m][scol + 1];
            // B fragment (band weights), computed from lane/k indices
            const int k0 = kbase, k1 = kbase + 1;
            v2f bb;
            if (o == 1)      { bb.x = 1.0f;                  bb.y = 1.0f; }
            else if (o == 0) { bb.x = (k0 > m) ? 1.0f : 0.0f; bb.y = (k1 > m) ? 1.0f : 0.0f; }
            else             { bb.x = (k0 < m) ? 1.0f : 0.0f; bb.y = (k1 < m) ? 1.0f : 0.0f; }
            d = __builtin_amdgcn_wmma_f32_16x16x4_f32(
                    false, a, false, bb, (short)0, d, false, false);
        }
    }

    // ---- Phase 4: fused weight + BCE + per-wave reduction
    const float inv_kk = 1.0f / 961.0f;
    float num = 0.0f, den = 0.0f;
    #pragma unroll
    for (int r = 0; r < 8; ++r) {
        const int M  = r + hi * 8;          // D layout: VGPR r, halves 8 apart
        const int hh = sr * 16 + M;         // row within 64x64 tile
        const int ww = sc * 16 + m;         // col within 64x64 tile
        const float pool = d[r] * inv_kk;
        const float t    = sT[hh + HALO][ww + HALO];
        const float l    = limg[(h0 + hh) * IMG_W + (w0 + ww)];
        const float wgt  = 1.0f + 5.0f * fabsf(pool - t);
        const float bce  = fmaxf(l, 0.0f) - l * t + log1pf(expf(-fabsf(l)));
        num += wgt * bce;
        den += wgt;
    }
    #pragma unroll
    for (int s = 16; s > 0; s >>= 1) {      // wave32 butterfly reduction
        num += __shfl_xor(num, s, 32);
        den += __shfl_xor(den, s, 32);
    }
    if (lane == 0) {
        atomicAdd(&acc[2 * b],     num);
        atomicAdd(&acc[2 * b + 1], den);
    }
}

__global__ void wbce_finalize(const float* __restrict__ acc, float* __restrict__ out) {
    const int t = threadIdx.x;              // 32 threads, one per image
    float r = acc[2 * t] / acc[2 * t + 1];
    #pragma unroll
    for (int s = 16; s > 0; s >>= 1) r += __shfl_xor(r, s, 32);
    if (t == 0) out[0] = r * (1.0f / (float)N_IMG);
}

extern "C" void kernel_launch(void* const* d_in, const int* in_sizes, int n_in,
                              void* d_out, int out_size, void* d_ws, size_t ws_size,
                              hipStream_t stream) {
    const float* logits  = (const float*)d_in[0];
    const float* targets = (const float*)d_in[1];
    float* acc = (float*)d_ws;              // 64 floats: {num,den} per image
    float* out = (float*)d_out;

    hipLaunchKernelGGL(wbce_init_acc, dim3(1), dim3(64), 0, stream, acc);
    hipLaunchKernelGGL(wbce_pool_kernel, dim3(N_IMG * 64), dim3(512), 0, stream,
                       logits, targets, acc);
    hipLaunchKernelGGL(wbce_finalize, dim3(1), dim3(32), 0, stream, acc, out);
}